// NLRidge_65962107732235
// MI455X (gfx1250) — compile-verified
//
#include <hip/hip_runtime.h>
#include <cmath>

typedef __attribute__((ext_vector_type(2))) float v2f;
typedef __attribute__((ext_vector_type(8))) float v8f;

#define IMG 384
#define PP 7
#define PS 49
#define WP 378            // IMG - PP + 1
#define LP (WP*WP)        // 142884 patch positions
#define NREF 96
#define NG (NREF*NREF)    // 9216 groups
#define WIN 37
#define WR 18
#define WW (WIN*WIN)      // 1369 offsets
#define TS 43             // candidate tile span: 37 offsets + 6 patch overhang
#define TLD 44
#define KA 18
#define KB 55
#define DVAL 30625.0f     // sigma^2 * C*P*P = 625*49
#define BIG 3.0e38f

// ---------------------------------------------------------------------------
// CDNA5 async global->LDS load (ASYNCcnt-tracked), with safe fallback.
// Probe result: builtin exists; params are (AS1 int*, AS3 int*, imm, imm).
// ---------------------------------------------------------------------------
#if defined(__has_builtin)
#if __has_builtin(__builtin_amdgcn_global_load_async_to_lds_b32)
#define HAVE_ASYNC_LDS 1
#endif
#endif
#ifndef HAVE_ASYNC_LDS
#define HAVE_ASYNC_LDS 0
#endif

typedef __attribute__((address_space(1))) int* gint_p;
typedef __attribute__((address_space(3))) int* lint_p;

__device__ __forceinline__ void async_wait0() {
#if defined(__has_builtin) && __has_builtin(__builtin_amdgcn_s_wait_asynccnt)
  __builtin_amdgcn_s_wait_asynccnt(0);
#else
  asm volatile("s_wait_asynccnt 0x0" ::: "memory");
#endif
}

// ---------------------------------------------------------------------------
// WMMA f32 16x16x4 wrapper (CDNA5): A 16x4 f32 (2 VGPR), B 4x16 f32 (2 VGPR),
// C/D 16x16 f32 (8 VGPR). 8-arg form: (neg_a, A, neg_b, B, c_mod, C, ra, rb)
// ---------------------------------------------------------------------------
__device__ __forceinline__ v8f wmma4(v2f a, v2f b, v8f c) {
  return __builtin_amdgcn_wmma_f32_16x16x4_f32(false, a, false, b, (short)0, c,
                                               false, false);
}

// ---------------------------------------------------------------------------
// utility kernels
// ---------------------------------------------------------------------------
__global__ void zero_kernel(float* __restrict__ p, int n) {
  int i = blockIdx.x * blockDim.x + threadIdx.x;
  if (i < n) p[i] = 0.f;
}

// ---------------------------------------------------------------------------
// block matching: one 128-thread block per reference patch.
// 43x43 candidate tile staged into LDS (async global->LDS when available),
// distances for all 1369 offsets into LDS, then K iterated block-argmins.
// ---------------------------------------------------------------------------
__global__ __launch_bounds__(128) void bm_kernel(const float* __restrict__ img,
                                                 int* __restrict__ idxOut,
                                                 int K) {
  __shared__ float tile[TS * TLD];
  __shared__ float refp[PS];
  __shared__ float dist[WW];
  __shared__ float rv[128];
  __shared__ int   ri[128];

  const int tid = threadIdx.x;
  const int g = blockIdx.x;
  const int tr = g / NREF, tc = g % NREF;
  const int rt = min(4 * tr, WP - 1), ct = min(4 * tc, WP - 1);
  const int r0 = rt - WR, c0 = ct - WR;

  // stage candidate pixel tile (clamped; out-of-range candidates never read)
  for (int t = tid; t < TS * TS; t += 128) {
    int i = t / TS, j = t % TS;
    int r = min(max(r0 + i, 0), IMG - 1);
    int c = min(max(c0 + j, 0), IMG - 1);
#if HAVE_ASYNC_LDS
    __builtin_amdgcn_global_load_async_to_lds_b32(
        (gint_p)(img + r * IMG + c), (lint_p)&tile[i * TLD + j], 0, 0);
#else
    tile[i * TLD + j] = img[r * IMG + c];
#endif
  }
#if HAVE_ASYNC_LDS
  async_wait0();  // per-wave drain of ASYNCcnt before the block barrier
#endif
  __syncthreads();

  if (tid < PS)
    refp[tid] = tile[(WR + tid / PP) * TLD + WR + tid % PP];
  __syncthreads();

  for (int o = tid; o < WW; o += 128) {
    int oi = o / WIN - WR, oj = o % WIN - WR;
    int r = rt + oi, c = ct + oj;
    float dv = BIG;
    if (r >= 0 && r < WP && c >= 0 && c < WP) {
      const float* tp = &tile[(oi + WR) * TLD + (oj + WR)];
      float s = 0.f;
#pragma unroll
      for (int di = 0; di < PP; ++di)
#pragma unroll
        for (int dj = 0; dj < PP; ++dj) {
          float diff = tp[di * TLD + dj] - refp[di * PP + dj] + 1e-6f;
          s += diff * diff;
        }
      dv = sqrtf(s);
    }
    dist[o] = dv;
  }
  __syncthreads();
  if (tid == 0) dist[WR * WIN + WR] = -BIG;  // self patch always first
  __syncthreads();

  for (int t = 0; t < K; ++t) {
    float bv = 3.4e38f;
    int bo = WW;
    for (int o = tid; o < WW; o += 128)
      if (dist[o] < bv) { bv = dist[o]; bo = o; }
    rv[tid] = bv; ri[tid] = bo;
    __syncthreads();
    for (int s = 64; s > 0; s >>= 1) {
      if (tid < s) {
        if (rv[tid + s] < rv[tid] ||
            (rv[tid + s] == rv[tid] && ri[tid + s] < ri[tid])) {
          rv[tid] = rv[tid + s]; ri[tid] = ri[tid + s];
        }
      }
      __syncthreads();
    }
    if (tid == 0) {
      int o = ri[0];
      int ir = rt + o / WIN - WR, ic = ct + o % WIN - WR;
      ir = min(max(ir, 0), WP - 1);
      ic = min(max(ic, 0), WP - 1);
      idxOut[g * K + t] = ir * WP + ic;
      dist[o] = BIG;
    }
    __syncthreads();
  }
}

// ---------------------------------------------------------------------------
// fused gather + denoise + aggregate, one 128-thread (4-wave) block per group.
//   stage1: Q = Y Y^T            theta = I - D*Qinv   X_hat = theta @ Y
//   stage2: Q = X X^T + D*I      theta = I - D*Qinv   X_hat = theta @ Y
// WMMA f32 16x16x4 used for both the Gram matrix and theta @ Y.
// ---------------------------------------------------------------------------
template <int K, bool STAGE2>
__global__ __launch_bounds__(128) void denoise_agg(
    const float* __restrict__ y, const float* __restrict__ x1,
    const int* __restrict__ idx, float* __restrict__ num,
    float* __restrict__ wsum) {
  constexpr int NT  = 128;
  constexpr int KP  = (K <= 32) ? 32 : 64;  // padded member count
  constexpr int LDY = 53;                   // 49 dims padded to 52, +1
  constexpr int LDQ = KP + 1;
  constexpr int MT  = KP / 16;

  __shared__ float Yg[KP * LDY];
  __shared__ float Xg[STAGE2 ? KP * LDY : 1];
  __shared__ float Qb[KP * LDQ];
  __shared__ float Zb[KP * LDQ];
  __shared__ float wts[KP];
  __shared__ int   pos[K];

  const int tid  = threadIdx.x;
  const int g    = blockIdx.x;
  const int lane = tid & 31, wave = tid >> 5;
  const int half = lane >> 4, lm = lane & 15;

  // zero-pad group buffers
  for (int t = tid; t < KP * LDY; t += NT) {
    Yg[t] = 0.f;
    if constexpr (STAGE2) Xg[t] = 0.f;
  }
  if (tid < K) pos[tid] = idx[g * K + tid];
  __syncthreads();

  // gather patch groups from global
  for (int t = tid; t < K * PS; t += NT) {
    int m = t / PS, d = t % PS;
    int p = pos[m];
    int r = p / WP + d / PP, c = p % WP + d % PP;
    Yg[m * LDY + d] = y[r * IMG + c];
    if constexpr (STAGE2) Xg[m * LDY + d] = x1[r * IMG + c];
  }
  __syncthreads();

  const float* G = STAGE2 ? (const float*)Xg : (const float*)Yg;

  // ---- Q = G G^T (WMMA). A rows = G, B = G^T (B[k][n] = G[n][k]). ----
  for (int tt = wave; tt < MT * MT; tt += 4) {
    int m0 = (tt / MT) * 16, n0 = (tt % MT) * 16;
    v8f acc = {};
    for (int k0 = 0; k0 < 52; k0 += 4) {
      v2f a, b;
      const float* pa = &G[(m0 + lm) * LDY + k0 + 2 * half];
      a.x = pa[0]; a.y = pa[1];
      const float* pb = &G[(n0 + lm) * LDY + k0 + 2 * half];
      b.x = pb[0]; b.y = pb[1];
      acc = wmma4(a, b, acc);
    }
#pragma unroll
    for (int vi = 0; vi < 8; ++vi)
      Qb[(m0 + vi + 8 * half) * LDQ + n0 + lm] = acc[vi];
  }
  __syncthreads();

  if constexpr (STAGE2) {
    for (int i = tid; i < K; i += NT) Qb[i * LDQ + i] += DVAL;
    __syncthreads();
  }

  // ---- Cholesky, in-place lower triangle of Qb ----
  for (int j = 0; j < K; ++j) {
    if (tid == 0) Qb[j * LDQ + j] = sqrtf(fmaxf(Qb[j * LDQ + j], 1e-12f));
    __syncthreads();
    float dj = Qb[j * LDQ + j];
    for (int i = j + 1 + tid; i < K; i += NT) Qb[i * LDQ + j] /= dj;
    __syncthreads();
    int rem = K - 1 - j;
    for (int t = tid; t < rem * rem; t += NT) {
      int i = j + 1 + t / rem, kk = j + 1 + t % rem;
      if (kk <= i) Qb[i * LDQ + kk] -= Qb[i * LDQ + j] * Qb[kk * LDQ + j];
    }
    __syncthreads();
  }

  // ---- Qinv = (L L^T)^-1 via two triangular solves, column-parallel ----
  for (int c0 = tid; c0 < K; c0 += NT) {
    for (int i = 0; i < K; ++i) {
      float s = (i == c0) ? 1.f : 0.f;
      for (int kk = 0; kk < i; ++kk) s -= Qb[i * LDQ + kk] * Zb[kk * LDQ + c0];
      Zb[i * LDQ + c0] = s / Qb[i * LDQ + i];
    }
    for (int i = K - 1; i >= 0; --i) {
      float s = Zb[i * LDQ + c0];
      for (int kk = i + 1; kk < K; ++kk)
        s -= Qb[kk * LDQ + i] * Zb[kk * LDQ + c0];
      Zb[i * LDQ + c0] = s / Qb[i * LDQ + i];
    }
  }
  __syncthreads();

  // ---- theta = I - D * Qinv (transposed; Qinv symmetric) into Qb, padded ----
  for (int t = tid; t < KP * KP; t += NT) {
    int i = t / KP, j = t % KP;
    float v = 0.f;
    if (i < K && j < K) v = ((i == j) ? 1.f : 0.f) - DVAL * Zb[j * LDQ + i];
    Qb[i * LDQ + j] = v;
  }
  __syncthreads();

  // ---- weights = 1 / clip(row-norm^2(theta), 1/K, 1) ----
  for (int i = tid; i < K; i += NT) {
    float s = 0.f;
    for (int j = 0; j < K; ++j) { float v = Qb[i * LDQ + j]; s += v * v; }
    wts[i] = 1.f / fminf(fmaxf(s, 1.f / (float)K), 1.f);
  }
  __syncthreads();

  // ---- X_hat = theta @ Y (WMMA) with fused weighted atomic scatter ----
  for (int tt = wave; tt < MT * 4; tt += 4) {
    int m0 = (tt / 4) * 16, n0 = (tt % 4) * 16;
    v8f acc = {};
    for (int k0 = 0; k0 < KP; k0 += 4) {
      v2f a, b;
      const float* pa = &Qb[(m0 + lm) * LDQ + k0 + 2 * half];
      a.x = pa[0]; a.y = pa[1];
      int nn = min(n0 + lm, 51);  // clamp; n>=49 outputs discarded below
      b.x = Yg[(k0 + 2 * half) * LDY + nn];
      b.y = Yg[(k0 + 2 * half + 1) * LDY + nn];
      acc = wmma4(a, b, acc);
    }
#pragma unroll
    for (int vi = 0; vi < 8; ++vi) {
      int m = m0 + vi + 8 * half, n = n0 + lm;
      if (m < K && n < PS)
        atomicAdd(&num[n * LP + pos[m]], acc[vi] * wts[m]);
    }
  }
  if (tid < K) atomicAdd(&wsum[pos[tid]], wts[tid]);
}

// ---------------------------------------------------------------------------
// overlap-add fold + normalize: out[r,c] = sum_ij num[ij][r-i,c-j] / sum wsum
// ---------------------------------------------------------------------------
__global__ void fold_div(const float* __restrict__ num,
                         const float* __restrict__ wsum,
                         float* __restrict__ out) {
  int p = blockIdx.x * blockDim.x + threadIdx.x;
  if (p >= IMG * IMG) return;
  int r = p / IMG, c = p % IMG;
  float ns = 0.f, ws = 0.f;
  for (int i = 0; i < PP; ++i) {
    int rr = r - i;
    if (rr < 0 || rr >= WP) continue;
    for (int j = 0; j < PP; ++j) {
      int cc = c - j;
      if (cc < 0 || cc >= WP) continue;
      ns += num[(i * PP + j) * LP + rr * WP + cc];
      ws += wsum[rr * WP + cc];
    }
  }
  out[p] = ns / fmaxf(ws, 1e-20f);
}

// ---------------------------------------------------------------------------
extern "C" void kernel_launch(void* const* d_in, const int* in_sizes, int n_in,
                              void* d_out, int out_size, void* d_ws,
                              size_t ws_size, hipStream_t stream) {
  (void)in_sizes; (void)n_in; (void)out_size; (void)ws_size;
  const float* y = (const float*)d_in[0];
  float* out = (float*)d_out;

  // workspace layout (floats): num[49*LP] | wsum[LP] | x1[IMG*IMG] | idx1 | idx2
  float* num  = (float*)d_ws;
  float* wsum = num + (size_t)PS * LP;
  float* x1   = wsum + LP;
  int*   idx1 = (int*)(x1 + IMG * IMG);
  int*   idx2 = idx1 + (size_t)NG * KA;

  const int nz = (PS + 1) * LP;  // num + wsum contiguous

  // ---- stage 1 ----
  zero_kernel<<<(nz + 255) / 256, 256, 0, stream>>>(num, nz);
  bm_kernel<<<NG, 128, 0, stream>>>(y, idx1, KA);
  denoise_agg<KA, false><<<NG, 128, 0, stream>>>(y, nullptr, idx1, num, wsum);
  fold_div<<<(IMG * IMG + 255) / 256, 256, 0, stream>>>(num, wsum, x1);

  // ---- stage 2 ----
  zero_kernel<<<(nz + 255) / 256, 256, 0, stream>>>(num, nz);
  bm_kernel<<<NG, 128, 0, stream>>>(x1, idx2, KB);
  denoise_agg<KB, true><<<NG, 128, 0, stream>>>(y, x1, idx2, num, wsum);
  fold_div<<<(IMG * IMG + 255) / 256, 256, 0, stream>>>(num, wsum, out);
}